// SpanRepresentation_84765474554683
// MI455X (gfx1250) — compile-verified
//
#include <hip/hip_runtime.h>

#define SEQ_LEN   8192
#define HIDDEN    1024
#define MAX_W     32
#define NEG_INF   (-1e9f)

// TDM-padded LDS tile layout: 1 pad dword after every 256 data dwords
// (pad_interval=7 -> 256 dwords, pad_amount=0 -> 1 dword).
// Row = 4 blocks of 257 floats = 1028 floats (4112 B): 16B-aligned row starts,
// 4-bank row skew => the 16 WMMA A-matrix lanes hit 16 distinct banks.
// element(p, h) lives at  p*ROWF + (h>>8)*BLK + (h&255)
#define BLK       257
#define ROWF      1028
#define E_FLOATS  (MAX_W * ROWF)        // 32896 floats = 128.5 KB gathered tile
#define AW_OFF    E_FLOATS              // attn_w copy (1024 floats, 16B aligned)
#define SC_OFF    (AW_OFF + HIDDEN)     // scores (32)
#define WT_OFF    (SC_OFF + MAX_W)      // weights (32)
#define LDS_FLOATS (WT_OFF + MAX_W)     // 33984 floats = 135936 B (2 blocks/WGP)

typedef __attribute__((ext_vector_type(2))) float    v2f;
typedef __attribute__((ext_vector_type(8))) float    v8f;
typedef __attribute__((ext_vector_type(4))) unsigned u32x4;
typedef __attribute__((ext_vector_type(8))) int      i32x8;
typedef __attribute__((ext_vector_type(4))) int      i32x4;

static __device__ __forceinline__ int imin(int a, int b) { return a < b ? a : b; }

// Issue one TDM gather: 16 rows x 1024 f32 from emb into LDS at lds_byte,
// using 16-bit row indices (gather mode), 1-dword pad per 256-dword block.
static __device__ __forceinline__ void tdm_gather16(const float* emb, int start,
                                                    int rbase, unsigned lds_byte)
{
    const unsigned long long ea = (unsigned long long)emb;

    u32x4 g0;
    g0[0] = 0x80000001u;                       // count=1 | gather_mode (16-bit idx)
    g0[1] = lds_byte;                          // LDS byte address of tile
    g0[2] = (unsigned)ea;                      // global_addr[31:0]
    g0[3] = (unsigned)((ea >> 32) & 0x01ffffffull) | 0x80000000u; // addr[56:32]|type=2

    i32x8 g1;
    g1[0] = (2 << 16)                          // data_size = 4 bytes
          | (1 << 20)                          // pad_enable (load)
          | (7 << 22)                          // pad_interval: 256 dwords
          | (0 << 25);                         // pad_amount: 1 dword
    g1[1] = (int)((HIDDEN & 0xffff) << 16);    // tensor_dim0[15:0] = 1024
    g1[2] = (int)((SEQ_LEN & 0xffff) << 16);   // tensor_dim1[15:0] = 8192 rows
    g1[3] = (int)((HIDDEN & 0xffff) << 16);    // tile_dim0 = 1024
    g1[4] = 16;                                // tile_dim1 = 16 valid indices
    g1[5] = HIDDEN;                            // tensor_dim0_stride[31:0] = 1024
    g1[6] = 0;
    g1[7] = 0;

    i32x4 g2, g3;                              // 16 packed 16-bit row indices
    #pragma unroll
    for (int w = 0; w < 4; ++w) {
        const int p0 = imin(start + rbase + 2 * w,     SEQ_LEN - 1);
        const int p1 = imin(start + rbase + 2 * w + 1, SEQ_LEN - 1);
        const int q0 = imin(start + rbase + 2 * w + 8, SEQ_LEN - 1);
        const int q1 = imin(start + rbase + 2 * w + 9, SEQ_LEN - 1);
        g2[w] = (p0 & 0xffff) | (p1 << 16);
        g3[w] = (q0 & 0xffff) | (q1 << 16);
    }

    const i32x8 gz = {};                       // extra operand (zero-filled, per probe)
    __builtin_amdgcn_tensor_load_to_lds(g0, g1, g2, g3, gz, 0);
}

extern "C" __global__ void __launch_bounds__(256)
span_repr_kernel(const float* __restrict__ emb,
                 const int*   __restrict__ starts,
                 const int*   __restrict__ ends,
                 const float* __restrict__ attn_w,
                 const float* __restrict__ attn_b,
                 float*       __restrict__ out)
{
    extern __shared__ float lds[];          // dynamic LDS: starts at LDS offset 0
    const int s    = blockIdx.x;
    const int t    = threadIdx.x;           // 0..255
    const int wave = t >> 5;                // 0..7
    const int lane = t & 31;

    const int start = __builtin_amdgcn_readfirstlane(starts[s]);
    const int end   = __builtin_amdgcn_readfirstlane(ends[s]);

    // ---- stage attn_w into LDS; zero score accumulators ---------------------
    ((float4*)(lds + AW_OFF))[t] = ((const float4*)attn_w)[t];   // 4 KB
    if (t < MAX_W) lds[SC_OFF + t] = 0.0f;

    // ---- TDM gather: 2 descriptors fetch the whole 128 KB span tile ---------
    if (wave == 0) {
        tdm_gather16(emb, start, 0,  0u);
        tdm_gather16(emb, start, 16, (unsigned)(16 * ROWF * 4));
        __builtin_amdgcn_s_wait_tensorcnt(0);
    }
    __syncthreads();

    // ---- scores via V_WMMA_F32_16X16X4_F32 ---------------------------------
    // Wave w: positions grp*16..grp*16+15 (grp = w>>2), K-slab (w&3)*256
    // (one TDM 256-dword block, so the pad offset is a wave constant).
    // A (16x4 f32 ISA layout): lanes 0-15 hold K={k0,k0+1}, lanes 16-31 K={k0+2,k0+3},
    // row M = lane%16.  B (4x16): every N column = same attn_w value, so the
    // D rows replicate the 16 partial scores.
    {
        const int   m    = lane & 15;
        const int   kh   = lane >> 4;            // 0 or 1
        const int   grp  = wave >> 2;            // 0 or 1
        const int   slab = wave & 3;             // which 256-dword block
        const float* Abase = lds + (grp * 16 + m) * ROWF + slab * BLK;
        const float* wv    = lds + AW_OFF + slab * 256;

        v8f c = {};
        #pragma unroll 8
        for (int kk = 0; kk < 64; ++kk) {
            const int k0 = kk * 4;
            v2f a, b;
            a.x = Abase[k0 + 2 * kh];
            a.y = Abase[k0 + 2 * kh + 1];
            b.x = wv[k0 + kh];
            b.y = wv[k0 + 2 + kh];
            c = __builtin_amdgcn_wmma_f32_16x16x4_f32(
                    /*neg_a=*/false, a, /*neg_b=*/false, b,
                    /*c_mod=*/(short)0, c, /*reuse_a=*/false, /*reuse_b=*/false);
        }
        // C/D layout: lane<16 holds M=0..7 in c[0..7]; lane>=16 holds M=8..15.
        if (m == 0) {
            const int mbase = grp * 16 + kh * 8;
            #pragma unroll
            for (int r = 0; r < 8; ++r)
                atomicAdd(&lds[SC_OFF + mbase + r], c[r]);   // ds_add_f32
        }
    }
    __syncthreads();

    // ---- masked softmax over the 32 positions (wave 0, wave32 shuffles) -----
    if (wave == 0) {
        const float bias = attn_b[0];
        float sc = lds[SC_OFF + lane] + bias;
        const bool valid = (start + lane) <= end;
        sc = valid ? sc : NEG_INF;
        float mx = sc;
        #pragma unroll
        for (int o = 16; o > 0; o >>= 1) mx = fmaxf(mx, __shfl_xor(mx, o, 32));
        const float e = __expf(sc - mx);
        float sum = e;
        #pragma unroll
        for (int o = 16; o > 0; o >>= 1) sum += __shfl_xor(sum, o, 32);
        lds[WT_OFF + lane] = e / sum;
    }
    __syncthreads();

    // ---- outputs ------------------------------------------------------------
    // Thread t owns h = {t, 256+t, 512+t, 768+t}: lane-consecutive LDS banks
    // (conflict-free) and fully coalesced 128B-per-instruction global stores.
    // start row == tile row 0; end row == tile row (end-start): free from LDS.
    float a0 = 0.f, a1 = 0.f, a2 = 0.f, a3 = 0.f;
    #pragma unroll 4
    for (int p = 0; p < MAX_W; ++p) {
        const float wp = lds[WT_OFF + p];
        const float* row = lds + p * ROWF + t;
        a0 += wp * row[0];
        a1 += wp * row[BLK];
        a2 += wp * row[2 * BLK];
        a3 += wp * row[3 * BLK];
    }

    const int widx = end - start;            // < MAX_W by construction
    const float* r0 = lds + t;               // start row
    const float* rw = lds + widx * ROWF + t; // end row
    float* outS = out + (size_t)s * (3 * HIDDEN);

    outS[t]                    = r0[0];
    outS[256 + t]              = r0[BLK];
    outS[512 + t]              = r0[2 * BLK];
    outS[768 + t]              = r0[3 * BLK];

    outS[HIDDEN + t]           = rw[0];
    outS[HIDDEN + 256 + t]     = rw[BLK];
    outS[HIDDEN + 512 + t]     = rw[2 * BLK];
    outS[HIDDEN + 768 + t]     = rw[3 * BLK];

    outS[2 * HIDDEN + t]       = a0;
    outS[2 * HIDDEN + 256 + t] = a1;
    outS[2 * HIDDEN + 512 + t] = a2;
    outS[2 * HIDDEN + 768 + t] = a3;
}

extern "C" void kernel_launch(void* const* d_in, const int* in_sizes, int n_in,
                              void* d_out, int out_size, void* d_ws, size_t ws_size,
                              hipStream_t stream)
{
    const float* emb    = (const float*)d_in[0];
    const int*   starts = (const int*)  d_in[1];
    const int*   ends   = (const int*)  d_in[2];
    const float* attn_w = (const float*)d_in[3];
    const float* attn_b = (const float*)d_in[4];
    float*       out    = (float*)d_out;

    const int num_spans = in_sizes[1];
    const size_t shmem  = (size_t)LDS_FLOATS * sizeof(float);   // ~133 KB dynamic LDS

    // Allow >64KB dynamic LDS (CDNA5 WGP has 320 KB). Idempotent; ignore errors.
    (void)hipFuncSetAttribute((const void*)span_repr_kernel,
                              hipFuncAttributeMaxDynamicSharedMemorySize,
                              (int)shmem);

    span_repr_kernel<<<dim3(num_spans), dim3(256), shmem, stream>>>(
        emb, starts, ends, attn_w, attn_b, out);
}